// ModelNew_3556232921879
// MI455X (gfx1250) — compile-verified
//
#include <hip/hip_runtime.h>
#include <hip/hip_bf16.h>

typedef float v2f __attribute__((ext_vector_type(2)));
typedef float v8f __attribute__((ext_vector_type(8)));

#define BATCH 128
#define CIN   3
#define DIN   18
#define HIN   34
#define WIN   34
#define COUT  16
#define DOUTD 16
#define HOUTD 32
#define WOUTD 32
#define KVOL  81          // 3*3*3*3
#define KSTEPS 20         // 20 full K=4 WMMA steps (k=0..79); k=80 handled as FMA tail
#define NTILES 1024       // 16*32*32 / 16
#define NPOS   16384      // D*H*W per (b,c)
#define CH_STRIDE (DIN*HIN*WIN)   // 20808
#define D_STRIDE  (HIN*WIN)       // 1156
#define OFF80 (2*CH_STRIDE + 2*D_STRIDE + 2*WIN + 2)   // k=80 -> ci=2,kd=2,kh=2,kw=2

__global__ __launch_bounds__(256)
void fused_conv3d_inorm_max(const float* __restrict__ x,
                            const float* __restrict__ cw,
                            const float* __restrict__ cb,
                            const float* __restrict__ mult,
                            float* __restrict__ out)
{
    __shared__ float s_sum[COUT];
    __shared__ float s_ssq[COUT];
    __shared__ float s_mean[COUT];
    __shared__ float s_inv[COUT];

    const int b    = blockIdx.x;
    const int tid  = threadIdx.x;
    const int lane = tid & 31;
    // Scalar (SGPR) wave id -> scalar tile loops, no EXEC-masked looping.
    const int wv   = __builtin_amdgcn_readfirstlane(tid >> 5);   // 0..7
    const int hl   = lane >> 4;    // lane half: 0 or 1
    const int n    = lane & 15;    // WMMA row/col index (M for A, N for B/C)

    if (tid < COUT) { s_sum[tid] = 0.f; s_ssq[tid] = 0.f; }
    __syncthreads();

    // ---- Preload A (weights 16x80) and per-lane im2col offsets (all valid, no guards).
    // A layout (16x4 f32): lane&15 = M, K = vg + 2*hl within each 4-wide step.
    v2f a_frag[KSTEPS];
    int koff[KSTEPS][2];
#pragma unroll
    for (int s = 0; s < KSTEPS; ++s) {
#pragma unroll
        for (int vg = 0; vg < 2; ++vg) {
            const int k = 4 * s + vg + 2 * hl;     // <= 79, always valid
            a_frag[s][vg] = cw[n * KVOL + k];      // conv_w[o=M][k]
            const int ci = k / 27;
            const int r  = k % 27;
            const int kd = r / 9;
            const int kh = (r % 9) / 3;
            const int kw = r % 3;
            koff[s][vg] = ci * CH_STRIDE + kd * D_STRIDE + kh * WIN + kw;
        }
    }

    // Per-lane channel constants: C/D VGPR v holds channel c = v + 8*hl.
    float bm[8], bb[8], wl[8];
#pragma unroll
    for (int v = 0; v < 8; ++v) {
        const int c = 8 * hl + v;
        bm[v] = mult[c];
        bb[v] = cb[c] * bm[v];                  // y = conv*m + b*m
        wl[v] = cw[c * KVOL + 80];              // k=80 tail weight
    }

    const float* xb = x + (long)b * (CIN * CH_STRIDE);

    // Implicit-GEMM conv for one 16(channel) x 16(position) tile via f32 WMMA,
    // plus rank-1 FMA tail for the 81st K element.
    auto conv_tile = [&](int t) -> v8f {
        const int d  = t >> 6;                  // scalar (t is SGPR-derived)
        const int h  = (t >> 1) & 31;
        const int w0 = (t & 1) << 4;
        const float* base = xb + (d * D_STRIDE + h * WIN + w0) + n;
        v8f acc = {0.f, 0.f, 0.f, 0.f, 0.f, 0.f, 0.f, 0.f};
#pragma unroll
        for (int s = 0; s < KSTEPS; ++s) {
            v2f bf;
            bf[0] = base[koff[s][0]];
            bf[1] = base[koff[s][1]];
            acc = __builtin_amdgcn_wmma_f32_16x16x4_f32(
                      false, a_frag[s], false, bf, (short)0, acc, false, false);
        }
        const float xv = base[OFF80];           // B[80][n]
#pragma unroll
        for (int v = 0; v < 8; ++v)
            acc[v] = fmaf(wl[v], xv, acc[v]);
        return acc;
    };

    // ---- Pass 1: accumulate per-channel sum / sum-of-squares of y.
    float psum[8] = {0.f, 0.f, 0.f, 0.f, 0.f, 0.f, 0.f, 0.f};
    float pssq[8] = {0.f, 0.f, 0.f, 0.f, 0.f, 0.f, 0.f, 0.f};
    for (int t = wv; t < NTILES; t += 8) {
        const int tn = t + 8;
        if (tn < NTILES) {                      // scalar condition
            const float* pf = xb + (tn >> 6) * D_STRIDE + ((tn >> 1) & 31) * WIN;
            __builtin_prefetch(pf, 0, 0);       // global_prefetch_b8
        }
        v8f acc = conv_tile(t);
#pragma unroll
        for (int v = 0; v < 8; ++v) {
            const float y = acc[v] * bm[v] + bb[v];
            psum[v] += y;
            pssq[v] += y * y;
        }
    }
#pragma unroll
    for (int v = 0; v < 8; ++v) {
        atomicAdd(&s_sum[8 * hl + v], psum[v]); // ds_add_f32
        atomicAdd(&s_ssq[8 * hl + v], pssq[v]);
    }
    __syncthreads();

    if (tid < COUT) {
        const float m   = s_sum[tid] * (1.f / NPOS);
        const float var = s_ssq[tid] * (1.f / NPOS) - m * m;
        s_mean[tid] = m;
        s_inv[tid]  = rsqrtf(var + 1e-5f);
    }
    __syncthreads();

    float mn[8], iv[8];
#pragma unroll
    for (int v = 0; v < 8; ++v) {
        mn[v] = s_mean[8 * hl + v];
        iv[v] = s_inv[8 * hl + v];
    }

    // ---- Pass 2: recompute conv, normalize, clamp, scale, max over channels.
    for (int t = wv; t < NTILES; t += 8) {
        v8f acc = conv_tile(t);
        float mx = -3.402823466e+38f;
#pragma unroll
        for (int v = 0; v < 8; ++v) {
            const float y = acc[v] * bm[v] + bb[v];
            float z = (y - mn[v]) * iv[v];
            z = fminf(fmaxf(z, -1.f), 1.f);
            z *= bm[v];
            mx = fmaxf(mx, z);
        }
        // combine channels 0-7 (lanes 0-15) with 8-15 (lanes 16-31): same position n
        mx = fmaxf(mx, __shfl_xor(mx, 16, 32));
        if (hl == 0) {
            const int d  = t >> 6;
            const int h  = (t >> 1) & 31;
            const int w0 = (t & 1) << 4;
            out[(((long)b * DOUTD + d) * HOUTD + h) * WOUTD + w0 + n] = mx;
        }
    }
}

extern "C" void kernel_launch(void* const* d_in, const int* in_sizes, int n_in,
                              void* d_out, int out_size, void* d_ws, size_t ws_size,
                              hipStream_t stream) {
    const float* x    = (const float*)d_in[0];
    const float* cw   = (const float*)d_in[1];
    const float* cb   = (const float*)d_in[2];
    const float* mult = (const float*)d_in[3];
    float* out = (float*)d_out;
    fused_conv3d_inorm_max<<<BATCH, 256, 0, stream>>>(x, cw, cb, mult, out);
}